// SinkhornCausalAttention_21328807592573
// MI455X (gfx1250) — compile-verified
//
#include <hip/hip_runtime.h>
#include <cstdint>
#include <cstddef>

// ---------------- problem constants ----------------
#define T_LEN   8192
#define DHDIM   64
#define NBUCK   64
#define BSZ     128
#define NBH     16
#define NROT    127                      // bsz - 1
#define NEG_MAX (-3.4028234663852886e38f)
#define ATTN_SCALE 0.044194173824159216f // 512^-0.5
#define KV_STRIDE  68                    // 64 + 4 pad: conflict-free, float4-aligned

typedef float v2f __attribute__((ext_vector_type(2)));
typedef float v8f __attribute__((ext_vector_type(8)));

static __device__ inline v8f vzero8() {
    v8f z = {0.f, 0.f, 0.f, 0.f, 0.f, 0.f, 0.f, 0.f};
    return z;
}

// ---------------------------------------------------------------------------
// Kernel 1: per-bucket sums of rotated K (for causal cum-average) + first row
// grid: bh*buckets blocks, 64 threads (one per dh)
// ---------------------------------------------------------------------------
__global__ __launch_bounds__(64) void bsum_kernel(const float* __restrict__ k,
                                                  float* __restrict__ bsum,
                                                  float* __restrict__ kfirst) {
    const int blk = blockIdx.x;          // 0..1023
    const int bh  = blk >> 6;
    const int u   = blk & 63;
    const int h   = bh & 7;
    const bool rot = (h >= 4);
    const int d   = threadIdx.x;
    const float* kb = k + (size_t)bh * T_LEN * DHDIM;

    float acc = 0.f, first = 0.f;
    for (int s = 0; s < BSZ; ++s) {
        int tt = u * BSZ + s;
        if (rot) tt = (tt + NROT) & (T_LEN - 1);
        float v = kb[(size_t)tt * DHDIM + d];
        if (s == 0) first = v;
        acc += v;
    }
    bsum[(size_t)blk * DHDIM + d]   = acc;
    kfirst[(size_t)blk * DHDIM + d] = first;
}

// ---------------------------------------------------------------------------
// Kernel 2: exclusive prefix over buckets -> x = [cumavg(k), k] at bucket head
// grid: 16 blocks, 64 threads (one per dh)
// ---------------------------------------------------------------------------
__global__ __launch_bounds__(64) void xbuild_kernel(const float* __restrict__ bsum,
                                                    const float* __restrict__ kfirst,
                                                    float* __restrict__ x) {
    const int bh = blockIdx.x;
    const int d  = threadIdx.x;
    float run = 0.f;
    for (int u = 0; u < NBUCK; ++u) {
        const size_t bu = (size_t)bh * NBUCK + u;
        float kf = kfirst[bu * DHDIM + d];
        float S  = run + kf;                       // cumsum up to index u*BSZ inclusive
        x[bu * (2 * DHDIM) + d]          = S / (float)(u * BSZ + 1);
        x[bu * (2 * DHDIM) + DHDIM + d]  = kf;
        run += bsum[bu * DHDIM + d];
    }
}

// ---------------------------------------------------------------------------
// Kernel 3: sort-net row -> (sel, val). R = tril(softmax(mask(leakyrelu(xW))))
// then top-1 one-hot * max  ==  single (bucket index, scale) per row.
// grid: bh*buckets blocks, 96 threads
// ---------------------------------------------------------------------------
__global__ __launch_bounds__(96) void rmat_kernel(const float* __restrict__ x,
                                                  const float* __restrict__ W,
                                                  int* __restrict__ sel,
                                                  float* __restrict__ val) {
    __shared__ float sh[NBUCK + 1];
    const int blk = blockIdx.x;
    const int bh  = blk >> 6;
    const int u   = blk & 63;
    const int h   = bh & 7;
    const int c   = threadIdx.x;

    if (c < NBUCK + 1) {
        const float* xr = x + ((size_t)bh * NBUCK + u) * (2 * DHDIM);
        const float* wc = W + (size_t)h * (2 * DHDIM) * (NBUCK + 1) + c;
        float acc = 0.f;
        for (int e = 0; e < 2 * DHDIM; ++e) acc += xr[e] * wc[(size_t)e * (NBUCK + 1)];
        float r = (acc > 0.f) ? acc : 0.01f * acc;     // leaky_relu(0.01)
        if (c > u) r = NEG_MAX;                        // mask cols > row
        sh[c] = r;
    }
    __syncthreads();
    if (c == 0) {
        float m = NEG_MAX;
        for (int i = 0; i <= NBUCK; ++i) m = fmaxf(m, sh[i]);
        float p[NBUCK + 1];
        float s = 0.f;
        for (int i = 0; i <= NBUCK; ++i) { p[i] = expf(sh[i] - m); s += p[i]; }
        float inv = 1.f / s;
        int   best = 0;
        float bv   = -1.f;
        for (int i = 0; i <= NBUCK; ++i) {
            float q = (i >= u) ? 0.f : p[i] * inv;     // tril(-1)
            if (q > bv) { bv = q; best = i; }          // first-max, like jnp.argmax
        }
        sel[blk] = best;                               // 0 => null bucket
        val[blk] = bv;
    }
}

// ---------------------------------------------------------------------------
// Kernel 4: attention per (bh, bucket). 8 waves x 16 rows. f32 WMMA 16x16x4.
// ---------------------------------------------------------------------------
__global__ __launch_bounds__(256) void attn_kernel(const float* __restrict__ q,
                                                   const float* __restrict__ k,
                                                   const float* __restrict__ v,
                                                   const float* __restrict__ nullk,
                                                   const float* __restrict__ nullv,
                                                   const int* __restrict__ sel,
                                                   const float* __restrict__ val,
                                                   float* __restrict__ out) {
    __shared__ float kv[BSZ * KV_STRIDE];      // 34 KB staging tile (Q / K / V)
    __shared__ float pbuf[8][16 * 17];         // per-wave P transpose buffer

    const int blk  = blockIdx.x;               // 0..1023
    const int bh   = blk >> 6;
    const int u    = blk & 63;
    const int h    = bh & 7;
    const bool rot = (h >= 4);
    const int tid  = threadIdx.x;
    const int wave = tid >> 5;
    const int lane = tid & 31;
    const int half = lane >> 4;                // 0: lanes 0-15, 1: lanes 16-31
    const int l16  = lane & 15;

    const float* qb = q + (size_t)bh * T_LEN * DHDIM;
    const float* kb = k + (size_t)bh * T_LEN * DHDIM;
    const float* vb = v + (size_t)bh * T_LEN * DHDIM;

    const int   s_sel = sel[blk];
    const float s_val = val[blk];

    // cooperative 128x64 tile load (rotated time indexing), optional scale/null row
    auto load_tile = [&](const float* src, int tbase, float sc, const float* nrow) {
        for (int i = tid; i < BSZ * 16; i += 256) {
            int row = i >> 4;
            int c4  = (i & 15) << 2;
            float4 x4;
            if (nrow) {
                x4 = *(const float4*)(nrow + c4);
            } else {
                int tt = tbase + row;
                if (rot) tt = (tt + NROT) & (T_LEN - 1);
                x4 = *(const float4*)(src + (size_t)tt * DHDIM + c4);
            }
            float4 y4 = make_float4(x4.x * sc, x4.y * sc, x4.z * sc, x4.w * sc);
            *(float4*)(&kv[row * KV_STRIDE + c4]) = y4;
        }
    };

    // ---- stage Q, pull A-fragments (16x4 f32 layout: lanes0-15 K=0,1; 16-31 K=2,3)
    load_tile(qb, u * BSZ, 1.f, nullptr);
    __syncthreads();
    v2f aq[16];
    {
        const int m = wave * 16 + l16;
#pragma unroll
        for (int s = 0; s < 16; ++s) {
            int c = 4 * s + 2 * half;
            aq[s].x = kv[m * KV_STRIDE + c];
            aq[s].y = kv[m * KV_STRIDE + c + 1];
        }
    }
    __syncthreads();

    v8f p[16];

    // ---- dots, cols 0..127: K_lo = val * (null row | bucket sel-1) ----
    if (s_sel == 0) load_tile(nullptr, 0, s_val, nullk + (size_t)h * DHDIM);
    else            load_tile(kb, (s_sel - 1) * BSZ, s_val, nullptr);
    __syncthreads();
#pragma unroll
    for (int nt = 0; nt < 8; ++nt) {
        v8f acc = vzero8();
#pragma unroll
        for (int s = 0; s < 16; ++s) {
            v2f b;
            int n = nt * 16 + l16;
            int c = 4 * s + 2 * half;
            b.x = kv[n * KV_STRIDE + c];
            b.y = kv[n * KV_STRIDE + c + 1];
            acc = __builtin_amdgcn_wmma_f32_16x16x4_f32(false, aq[s], false, b,
                                                        (short)0, acc, false, false);
        }
        p[nt] = acc;
    }
    __syncthreads();

    // ---- dots, cols 128..255: K_hi = local bucket ----
    load_tile(kb, u * BSZ, 1.f, nullptr);
    __syncthreads();
#pragma unroll
    for (int nt = 0; nt < 8; ++nt) {
        v8f acc = vzero8();
#pragma unroll
        for (int s = 0; s < 16; ++s) {
            v2f b;
            int n = nt * 16 + l16;
            int c = 4 * s + 2 * half;
            b.x = kv[n * KV_STRIDE + c];
            b.y = kv[n * KV_STRIDE + c + 1];
            acc = __builtin_amdgcn_wmma_f32_16x16x4_f32(false, aq[s], false, b,
                                                        (short)0, acc, false, false);
        }
        p[8 + nt] = acc;
    }
    __syncthreads();

    // ---- scale + causal mask + softmax over 256 cols (rows in registers) ----
    const bool special = rot && (u == NBUCK - 1);
#pragma unroll
    for (int t = 0; t < 16; ++t) {
        int col = t * 16 + l16;
#pragma unroll
        for (int r = 0; r < 8; ++r) {
            int m = wave * 16 + r + 8 * half;          // C-layout row
            bool ok;
            if (special) ok = (col <= BSZ) ? (m == 0) : ((col - BSZ) <= m);
            else         ok = (col < BSZ) || ((col - BSZ) <= m);
            p[t][r] = ok ? p[t][r] * ATTN_SCALE : NEG_MAX;
        }
    }
    float mx[8], sm[8];
#pragma unroll
    for (int r = 0; r < 8; ++r) {
        float m_ = NEG_MAX;
#pragma unroll
        for (int t = 0; t < 16; ++t) m_ = fmaxf(m_, p[t][r]);
#pragma unroll
        for (int off = 1; off < 16; off <<= 1) m_ = fmaxf(m_, __shfl_xor(m_, off, 32));
        mx[r] = m_;
    }
#pragma unroll
    for (int r = 0; r < 8; ++r) {
        float s_ = 0.f;
#pragma unroll
        for (int t = 0; t < 16; ++t) {
            float e = expf(p[t][r] - mx[r]);
            p[t][r] = e;
            s_ += e;
        }
#pragma unroll
        for (int off = 1; off < 16; off <<= 1) s_ += __shfl_xor(s_, off, 32);
        sm[r] = s_;
    }
#pragma unroll
    for (int r = 0; r < 8; ++r) {
        float inv = 1.f / sm[r];
#pragma unroll
        for (int t = 0; t < 16; ++t) p[t][r] *= inv;
    }

    // ---- out = P @ V'  (C-layout P -> A-layout via wave-private LDS tile) ----
    v8f o[4];
#pragma unroll
    for (int i = 0; i < 4; ++i) o[i] = vzero8();

    // V_lo
    if (s_sel == 0) load_tile(nullptr, 0, s_val, nullv + (size_t)h * DHDIM);
    else            load_tile(vb, (s_sel - 1) * BSZ, s_val, nullptr);
    __syncthreads();
#pragma unroll
    for (int jt = 0; jt < 8; ++jt) {
#pragma unroll
        for (int r = 0; r < 8; ++r)
            pbuf[wave][(r + 8 * half) * 17 + l16] = p[jt][r];   // same-wave LDS: in-order
#pragma unroll
        for (int sl = 0; sl < 4; ++sl) {
            v2f a;
            a.x = pbuf[wave][l16 * 17 + 4 * sl + 2 * half];
            a.y = pbuf[wave][l16 * 17 + 4 * sl + 2 * half + 1];
#pragma unroll
            for (int dt = 0; dt < 4; ++dt) {
                v2f b;
                int jrow = jt * 16 + 4 * sl + 2 * half;
                b.x = kv[jrow * KV_STRIDE + dt * 16 + l16];
                b.y = kv[(jrow + 1) * KV_STRIDE + dt * 16 + l16];
                o[dt] = __builtin_amdgcn_wmma_f32_16x16x4_f32(false, a, false, b,
                                                              (short)0, o[dt], false, false);
            }
        }
    }
    __syncthreads();

    // V_hi
    load_tile(vb, u * BSZ, 1.f, nullptr);
    __syncthreads();
#pragma unroll
    for (int jt = 8; jt < 16; ++jt) {
#pragma unroll
        for (int r = 0; r < 8; ++r)
            pbuf[wave][(r + 8 * half) * 17 + l16] = p[jt][r];
#pragma unroll
        for (int sl = 0; sl < 4; ++sl) {
            v2f a;
            a.x = pbuf[wave][l16 * 17 + 4 * sl + 2 * half];
            a.y = pbuf[wave][l16 * 17 + 4 * sl + 2 * half + 1];
#pragma unroll
            for (int dt = 0; dt < 4; ++dt) {
                v2f b;
                int jrow = (jt - 8) * 16 + 4 * sl + 2 * half;
                b.x = kv[jrow * KV_STRIDE + dt * 16 + l16];
                b.y = kv[(jrow + 1) * KV_STRIDE + dt * 16 + l16];
                o[dt] = __builtin_amdgcn_wmma_f32_16x16x4_f32(false, a, false, b,
                                                              (short)0, o[dt], false, false);
            }
        }
    }

    // ---- store (undo head rotation on write) ----
#pragma unroll
    for (int dt = 0; dt < 4; ++dt) {
#pragma unroll
        for (int r = 0; r < 8; ++r) {
            int m  = wave * 16 + r + 8 * half;
            int tt = u * BSZ + m;
            if (rot) tt = (tt + NROT) & (T_LEN - 1);   // out[(s+n)%t] = out_rot[s]
            out[(size_t)bh * T_LEN * DHDIM + (size_t)tt * DHDIM + dt * 16 + l16] = o[dt][r];
        }
    }
}

// ---------------------------------------------------------------------------
extern "C" void kernel_launch(void* const* d_in, const int* in_sizes, int n_in,
                              void* d_out, int out_size, void* d_ws, size_t ws_size,
                              hipStream_t stream) {
    const float* q     = (const float*)d_in[0];
    const float* k     = (const float*)d_in[1];
    const float* v     = (const float*)d_in[2];
    const float* nullk = (const float*)d_in[3];
    const float* nullv = (const float*)d_in[4];
    const float* W     = (const float*)d_in[5];
    float* out = (float*)d_out;

    // workspace layout
    float* bsum   = (float*)d_ws;                         // 16*64*64
    float* kfirst = bsum   + NBH * NBUCK * DHDIM;         // 16*64*64
    float* x      = kfirst + NBH * NBUCK * DHDIM;         // 16*64*128
    int*   sel    = (int*)  (x + NBH * NBUCK * 2 * DHDIM);// 1024
    float* val    = (float*)(sel + NBH * NBUCK);          // 1024

    bsum_kernel<<<NBH * NBUCK, 64, 0, stream>>>(k, bsum, kfirst);
    xbuild_kernel<<<NBH, 64, 0, stream>>>(bsum, kfirst, x);
    rmat_kernel<<<NBH * NBUCK, 96, 0, stream>>>(x, W, sel, val);
    attn_kernel<<<NBH * NBUCK, 256, 0, stream>>>(q, k, v, nullk, nullv, sel, val, out);
    (void)in_sizes; (void)n_in; (void)out_size; (void)ws_size;
}